// Attn_61366492725368
// MI455X (gfx1250) — compile-verified
//
#include <hip/hip_runtime.h>
#include <math.h>

typedef __attribute__((ext_vector_type(2))) float v2f;
typedef __attribute__((ext_vector_type(8))) float v8f;

#define B_ 64
#define T_ 2048
#define H_ 512
#define Q_ 512

// -----------------------------------------------------------------------------
// Kernel 1: V[b,h] = sum_q query[b,q] * W[q,h]   ([64,512] = [64,512]x[512,512])
// One wave per 16x16 output tile, V_WMMA_F32_16X16X4_F32 over K=Q in steps of 4.
//
// A (16x4 f32) lane layout (ISA 7.12.2): lanes 0-15 hold M=lane, K={0,1} in
// v[0],v[1]; lanes 16-31 hold M=lane-16, K={2,3}.  B (4x16) mirrors with rows
// striped across lanes within a VGPR.  C/D: VGPR i holds M=i (lanes 0-15) and
// M=i+8 (lanes 16-31), N=lane&15.
// -----------------------------------------------------------------------------
__global__ __launch_bounds__(32) void proj_gemm(const float* __restrict__ query,
                                                const float* __restrict__ W,
                                                float* __restrict__ V) {
    const int lane = threadIdx.x & 31;
    const int h0   = blockIdx.x * 16;   // N tile over H
    const int b0   = blockIdx.y * 16;   // M tile over B
    const int row  = lane & 15;
    const int ksel = (lane >> 4) << 1;  // 0 for lanes 0-15, 2 for lanes 16-31

    const float* qrow = query + (size_t)(b0 + row) * Q_;

    v8f acc = {};
    #pragma unroll 4
    for (int q = 0; q < Q_; q += 4) {
        v2f a, bf;
        // A fragment: query[b0+row, q+ksel .. q+ksel+1]  (contiguous -> b64 load)
        a.x = qrow[q + ksel];
        a.y = qrow[q + ksel + 1];
        // B fragment: W[q+ksel .. q+ksel+1, h0+row]
        bf.x = W[(size_t)(q + ksel)     * H_ + h0 + row];
        bf.y = W[(size_t)(q + ksel + 1) * H_ + h0 + row];
        acc = __builtin_amdgcn_wmma_f32_16x16x4_f32(false, a, false, bf,
                                                    (short)0, acc, false, false);
    }

    const int col   = lane & 15;
    const int rbase = (lane >> 4) * 8;
    #pragma unroll
    for (int i = 0; i < 8; ++i)
        V[(size_t)(b0 + rbase + i) * H_ + h0 + col] = acc[i];
}

// -----------------------------------------------------------------------------
// Kernel 2: fused energies + masked online softmax + context, one block per b.
// 512 threads = 16 wave32s.  Wave w handles t = w, w+16, ...  Lane l owns
// h = 16*l .. 16*l+15 (4x float4 loads per row).  enc is read exactly once.
// Bias is dropped: it adds a per-b constant to every logit, and softmax is
// shift-invariant, so the reference output is independent of it.
// -----------------------------------------------------------------------------
__global__ __launch_bounds__(512) void attn_ctx(const float* __restrict__ enc,
                                                const float* __restrict__ V,
                                                const int* __restrict__ tgt,
                                                float* __restrict__ out) {
    const int b    = blockIdx.x;
    const int tid  = threadIdx.x;
    const int lane = tid & 31;
    const int wave = tid >> 5;

    int lim = tgt[b] + 1;            // valid positions: t <= tgt[b]
    if (lim > T_) lim = T_;

    // per-lane slice of V[b]
    float vloc[16];
    const float* vb = V + (size_t)b * H_ + lane * 16;
    #pragma unroll
    for (int j = 0; j < 16; ++j) vloc[j] = vb[j];

    float m = -INFINITY;
    float s = 0.0f;
    float ctx[16];
    #pragma unroll
    for (int j = 0; j < 16; ++j) ctx[j] = 0.0f;

    const float* encb = enc + (size_t)b * T_ * H_;

    for (int t = wave; t < lim; t += 16) {
        const float* row = encb + (size_t)t * H_ + lane * 16;
        if (t + 16 < lim) __builtin_prefetch(row + 16 * H_, 0, 1);

        float4 e0 = ((const float4*)row)[0];
        float4 e1 = ((const float4*)row)[1];
        float4 e2 = ((const float4*)row)[2];
        float4 e3 = ((const float4*)row)[3];
        float ev[16] = {e0.x, e0.y, e0.z, e0.w, e1.x, e1.y, e1.z, e1.w,
                        e2.x, e2.y, e2.z, e2.w, e3.x, e3.y, e3.z, e3.w};

        // partial dot, then wave32 butterfly reduce -> all lanes get energy
        float d = 0.0f;
        #pragma unroll
        for (int j = 0; j < 16; ++j) d = fmaf(vloc[j], ev[j], d);
        #pragma unroll
        for (int off = 16; off >= 1; off >>= 1) d += __shfl_xor(d, off, 32);

        // online softmax update (flash-style)
        float nm    = fmaxf(m, d);
        float scale = __expf(m - nm);   // 0 on first valid row (m = -inf)
        float p     = __expf(d - nm);
        s = s * scale + p;
        #pragma unroll
        for (int j = 0; j < 16; ++j) ctx[j] = fmaf(ctx[j], scale, p * ev[j]);
        m = nm;
    }

    // ---- merge the 16 per-wave partials ----
    __shared__ float sm[16];
    __shared__ float ss[16];
    __shared__ float sctx[H_];

    if (lane == 0) { sm[wave] = m; ss[wave] = s; }
    sctx[tid] = 0.0f;                 // 512 threads cover H_ exactly
    __syncthreads();

    float gm = -INFINITY;
    #pragma unroll
    for (int w = 0; w < 16; ++w) gm = fmaxf(gm, sm[w]);
    float gs = 0.0f;
    #pragma unroll
    for (int w = 0; w < 16; ++w) gs += ss[w] * __expf(sm[w] - gm);

    const float factor = __expf(m - gm);   // 0 for waves that saw no rows
    #pragma unroll
    for (int j = 0; j < 16; ++j)
        atomicAdd(&sctx[lane * 16 + j], ctx[j] * factor);   // ds_add_f32
    __syncthreads();

    out[(size_t)b * H_ + tid] = sctx[tid] / gs;
}

// -----------------------------------------------------------------------------
extern "C" void kernel_launch(void* const* d_in, const int* in_sizes, int n_in,
                              void* d_out, int out_size, void* d_ws, size_t ws_size,
                              hipStream_t stream) {
    const float* query = (const float*)d_in[0];   // [B,Q]
    const float* enc   = (const float*)d_in[1];   // [B,T,H]
    const float* W     = (const float*)d_in[2];   // [Q,H]
    // d_in[3] = bias [Q]  -- provably cancels under softmax shift-invariance
    const int*   tgt   = (const int*)d_in[4];     // [B]
    float*       outp  = (float*)d_out;           // [B,H]

    float* V = (float*)d_ws;                      // [B,H] = 128 KB scratch

    proj_gemm<<<dim3(H_ / 16, B_ / 16), 32, 0, stream>>>(query, W, V);
    attn_ctx<<<B_, 512, 0, stream>>>(enc, V, tgt, outp);
}